// RNNModelScratch_15178414424713
// MI455X (gfx1250) — compile-verified
//
#include <hip/hip_runtime.h>
#include <hip/hip_bf16.h>
#include <math.h>

// Problem constants (from reference)
#define VOCAB 32000
#define HD    256
#define BD    32
#define TD    128
#define VH    32256       // VOCAB + HD, row stride of Wz/Wr/Wh
#define WH_STRIDE 260     // padded LDS row stride: %4==0 (float4 ok), %64==4 (2-cycle b128, no worse)

typedef float v2f __attribute__((ext_vector_type(2)));
typedef float v8f __attribute__((ext_vector_type(8)));

__device__ __forceinline__ float sigmoidf_(float x) {
    return 1.0f / (1.0f + expf(-x));
}

// ---------------------------------------------------------------------------
// Phase 1: GRU recurrence. One workgroup per batch element (batches are
// independent -> no inter-WG sync). H state in LDS. 512 threads = 16 waves.
// CDNA5 perk: 320 KB LDS per WGP -> cache the entire Wh_h gate matrix
// (256x256 fp32, 256 KB) in LDS once; only Wz_h/Wr_h stream from L2 per step.
//   Preload: cooperative copy of Wh_h into padded LDS
//   Phase A: grp 0 computes z-dots, grp 1 computes r-dots (full K=256 each)
//   Combine: z = sig(.), rH = sig(.)*Hs
//   Phase B: both grps split the h-candidate dot over K halves (LDS weights)
//   Final : Hn = z*tanh(.) + (1-z)*Hs ; write H_all[t] (and Hf at t=T-1)
// ---------------------------------------------------------------------------
__global__ __launch_bounds__(512) void gru_recurrence(
    const int*   __restrict__ X,  const float* __restrict__ H0,
    const float* __restrict__ Wz, const float* __restrict__ bz,
    const float* __restrict__ Wr, const float* __restrict__ br,
    const float* __restrict__ Wh, const float* __restrict__ bh,
    float* __restrict__ Hall, float* __restrict__ Hf)
{
    __shared__ __align__(16) float WhL[HD * WH_STRIDE];  // 266,240 B
    __shared__ __align__(16) float Hs[HD];
    __shared__ __align__(16) float rH[HD];
    __shared__ float accZR[2][HD];
    __shared__ float accH[2][HD];

    const int tid = threadIdx.x;
    const int b   = blockIdx.x;          // batch index
    const int grp = tid >> 8;            // 0 or 1
    const int n   = tid & (HD - 1);      // hidden unit this thread owns

    // Phase-A role: grp==0 -> z gate, grp==1 -> r gate
    const float*  rowA   = ((grp == 0) ? Wz : Wr) + (size_t)n * VH;
    const float4* rowA_h = (const float4*)(rowA + VOCAB);   // 16B aligned: (n*32256+32000)%4==0
    const float   biasA  = (grp == 0) ? bz[n] : br[n];

    // Phase-B role: both groups split the h-gate dot over K (128 each)
    const float*  rowH  = Wh + (size_t)n * VH;              // for the x-gather only
    const float   biasH = bh[n];

    // ---- Preload Wh_h (recurrent part of Wh) into LDS, padded rows ----
    {
        const int prow  = tid >> 1;           // 0..255
        const int phalf = (tid & 1) * 128;    // 0 or 128
        const float4* src = (const float4*)(Wh + (size_t)prow * VH + VOCAB + phalf);
        float4*       dst = (float4*)(&WhL[prow * WH_STRIDE + phalf]);
        #pragma unroll 8
        for (int k4 = 0; k4 < 128 / 4; ++k4) dst[k4] = src[k4];
    }
    if (tid < HD) Hs[tid] = H0[(size_t)b * HD + tid];
    __syncthreads();

    const float4* rowH_lds = (const float4*)(&WhL[n * WH_STRIDE + grp * 128]);

    for (int t = 0; t < TD; ++t) {
        const int x = X[b * TD + t];     // uniform within the block

        // ---- Phase A: z and r dots over Hs (weights stream from L2) ----
        {
            const float g = rowA[x];     // embedding-gather term Wg[n, x]
            const float4* hs4 = (const float4*)Hs;
            float sx = 0.f, sy = 0.f, sz = 0.f, sw = 0.f;
            #pragma unroll 8
            for (int k4 = 0; k4 < HD / 4; ++k4) {
                float4 w = rowA_h[k4];
                float4 h = hs4[k4];
                sx = fmaf(w.x, h.x, sx);
                sy = fmaf(w.y, h.y, sy);
                sz = fmaf(w.z, h.z, sz);
                sw = fmaf(w.w, h.w, sw);
            }
            accZR[grp][n] = (sx + sy) + (sz + sw) + g + biasA;
        }
        __syncthreads();

        float hsn = 0.f, zn = 0.f;
        if (tid < HD) {
            hsn = Hs[tid];
            zn  = sigmoidf_(accZR[0][tid]);
            float r = sigmoidf_(accZR[1][tid]);
            rH[tid] = r * hsn;           // (r ⊙ Hs) feeds the h-gate matmul
        }
        __syncthreads();

        // ---- Phase B: h-candidate dot over rH, K split in halves (LDS) ----
        {
            const float4* rh4 = (const float4*)rH + grp * (HD / 8);
            float sx = 0.f, sy = 0.f, sz = 0.f, sw = 0.f;
            #pragma unroll 8
            for (int k4 = 0; k4 < HD / 8; ++k4) {
                float4 w = rowH_lds[k4];
                float4 h = rh4[k4];
                sx = fmaf(w.x, h.x, sx);
                sy = fmaf(w.y, h.y, sy);
                sz = fmaf(w.z, h.z, sz);
                sw = fmaf(w.w, h.w, sw);
            }
            float part = (sx + sy) + (sz + sw);
            if (grp == 0) part += rowH[x] + biasH;   // x-gather stays global (3 dwords/step)
            accH[grp][n] = part;
        }
        __syncthreads();

        if (tid < HD) {
            float h  = tanhf(accH[0][tid] + accH[1][tid]);
            float hn = zn * h + (1.0f - zn) * hsn;
            Hall[((size_t)t * BD + b) * HD + tid] = hn;   // row m = t*B + b
            if (t == TD - 1) Hf[(size_t)b * HD + tid] = hn;
            Hs[tid] = hn;
        }
        __syncthreads();
    }
}

// ---------------------------------------------------------------------------
// Phase 2: Y(4096 x 32000) = Hall(4096 x 256) @ Wo^T + bo, exact fp32 via
// V_WMMA_F32_16X16X4_F32. One wave owns a 32x32 output tile (2x2 WMMA tiles,
// each fragment load feeds two WMMAs). Store-bandwidth-bound -> NT stores.
// Fragment layouts (f32):
//   A 16x4 : element (m,k) -> lane = (m&15) + 16*(k>>1), vgpr = k&1
//   B 4x16 : element (k,n) -> lane = (n&15) + 16*(k>>1), vgpr = k&1
//   C/D    : vgpr r, lane L -> (M = r + 8*(L>>4), N = L&15)
// -> per-lane A/B fragments are contiguous 8-byte loads.
// ---------------------------------------------------------------------------
__global__ __launch_bounds__(256) void out_gemm(
    const float* __restrict__ Hall, const float* __restrict__ Wo,
    const float* __restrict__ bo,   float* __restrict__ Y)
{
    const int lane  = threadIdx.x & 31;
    const int wid   = threadIdx.x >> 5;
    const int gw    = blockIdx.x * 8 + wid;   // global wave id, 0..127999
    const int mtile = gw & 127;               // 128 m-tiles of 32 rows
    const int ntile = gw >> 7;                // 1000 n-tiles of 32 cols
    const int m0    = mtile * 32;
    const int n0    = ntile * 32;
    const int l     = lane & 15;
    const int half  = lane >> 4;

    const float b0v = bo[n0 + l];
    const float b1v = bo[n0 + 16 + l];
    v8f c00, c01, c10, c11;
    #pragma unroll
    for (int i = 0; i < 8; ++i) { c00[i] = b0v; c01[i] = b1v; c10[i] = b0v; c11[i] = b1v; }

    const float* A0 = Hall + (size_t)(m0 + l)      * HD + 2 * half;
    const float* A1 = Hall + (size_t)(m0 + 16 + l) * HD + 2 * half;
    const float* B0 = Wo   + (size_t)(n0 + l)      * HD + 2 * half;
    const float* B1 = Wo   + (size_t)(n0 + 16 + l) * HD + 2 * half;

    #pragma unroll 4
    for (int k = 0; k < HD; k += 4) {
        v2f a0 = *(const v2f*)(A0 + k);
        v2f a1 = *(const v2f*)(A1 + k);
        v2f b0 = *(const v2f*)(B0 + k);
        v2f b1 = *(const v2f*)(B1 + k);
        c00 = __builtin_amdgcn_wmma_f32_16x16x4_f32(false, a0, false, b0, (short)0, c00, false, false);
        c01 = __builtin_amdgcn_wmma_f32_16x16x4_f32(false, a0, false, b1, (short)0, c01, false, false);
        c10 = __builtin_amdgcn_wmma_f32_16x16x4_f32(false, a1, false, b0, (short)0, c10, false, false);
        c11 = __builtin_amdgcn_wmma_f32_16x16x4_f32(false, a1, false, b1, (short)0, c11, false, false);
    }

    #pragma unroll
    for (int r = 0; r < 8; ++r) {
        const int mrow = m0 + r + 8 * half;
        float* y0 = Y + (size_t)mrow * VOCAB;
        float* y1 = Y + (size_t)(mrow + 16) * VOCAB;
        __builtin_nontemporal_store(c00[r], y0 + n0 + l);
        __builtin_nontemporal_store(c01[r], y0 + n0 + 16 + l);
        __builtin_nontemporal_store(c10[r], y1 + n0 + l);
        __builtin_nontemporal_store(c11[r], y1 + n0 + 16 + l);
    }
}

extern "C" void kernel_launch(void* const* d_in, const int* in_sizes, int n_in,
                              void* d_out, int out_size, void* d_ws, size_t ws_size,
                              hipStream_t stream) {
    (void)in_sizes; (void)n_in; (void)out_size; (void)ws_size;
    const int*   X  = (const int*)  d_in[0];
    const float* H0 = (const float*)d_in[1];
    const float* Wz = (const float*)d_in[2];
    const float* bz = (const float*)d_in[3];
    const float* Wr = (const float*)d_in[4];
    const float* br = (const float*)d_in[5];
    const float* Wh = (const float*)d_in[6];
    const float* bh = (const float*)d_in[7];
    const float* Wo = (const float*)d_in[8];
    const float* bo = (const float*)d_in[9];

    float* Y    = (float*)d_out;                       // (T*B, V)
    float* Hf   = Y + (size_t)TD * BD * VOCAB;         // (B, H) tail
    float* Hall = (float*)d_ws;                        // (T*B, H) scratch, 4 MB

    gru_recurrence<<<BD, 512, 0, stream>>>(X, H0, Wz, bz, Wr, br, Wh, bh, Hall, Hf);

    const int tiles = (TD * BD / 32) * (VOCAB / 32);   // 128 * 1000
    out_gemm<<<tiles / 8, 256, 0, stream>>>(Hall, Wo, bo, Y);
}